// ActionNet_1915555414503
// MI455X (gfx1250) — compile-verified
//
#include <hip/hip_runtime.h>
#include <hip/hip_bf16.h>
#include <stdint.h>

typedef __attribute__((ext_vector_type(2))) float v2f;
typedef __attribute__((ext_vector_type(8))) float v8f;

#define DIN  128
#define HID  64
#define DOUT 2

// ---------------------------------------------------------------------------
// Kernel 1: layer-0 edge scatter.  One wave32 per edge; each lane handles 4
// contiguous features (float4).  msg = x[src] * env_attr, atomic-added into
// agg0[dst]; lane 0 bumps the degree count (shared by both layers).
// Edge id is made wave-uniform via readfirstlane so the edge_index fetches
// lower to scalar loads (one per wave, not 32 lanes).
// ---------------------------------------------------------------------------
__global__ void edge_scatter_l0(const float* __restrict__ x,
                                const int*   __restrict__ ei,
                                const float* __restrict__ eattr,
                                float* __restrict__ agg,
                                float* __restrict__ cnt,
                                int E)
{
    int wib  = __builtin_amdgcn_readfirstlane(threadIdx.x >> 5);
    int e    = blockIdx.x * (blockDim.x >> 5) + wib;
    int lane = threadIdx.x & 31;
    if (e >= E) return;
    int s = ei[e];
    int d = ei[(size_t)E + e];
    float4 xv = ((const float4*)(x + (size_t)s * DIN))[lane];
    float4 ev = ((const float4*)(eattr + (size_t)e * DIN))[lane];
    float* a = agg + (size_t)d * DIN + lane * 4;
    atomicAdd(a + 0, xv.x * ev.x);
    atomicAdd(a + 1, xv.y * ev.y);
    atomicAdd(a + 2, xv.z * ev.z);
    atomicAdd(a + 3, xv.w * ev.w);
    if (lane == 0) atomicAdd(cnt + d, 1.0f);
}

// ---------------------------------------------------------------------------
// Kernel 2: layer-0 node GEMM with V_WMMA_F32_16X16X4_F32.
// h = relu(concat([x, agg0/cnt]) @ W0 + b0),  [N,256]@[256,64].
// One wave per 32-node tile (two 16-row A fragments sharing each B fragment:
// 8 WMMAs per K-step vs 4 B loads).  W0 is staged into LDS K-pair-interleaved
// (lds[p][n] = {W0[2p][n], W0[2p+1][n]}) so each lane's B fragment is one
// contiguous float2 -> single ds_load_b64, no register shuffles.
// A-fragment layout (16x4 f32): lane l, row m=l&15, K pair koff=(l>>4)*2.
// C/D layout: VGPR v -> row (v + (l>>4)*8), col l&15.
// ---------------------------------------------------------------------------
__global__ void node_gemm_l0(const float* __restrict__ x,
                             const float* __restrict__ agg,
                             const float* __restrict__ cnt,
                             const float* __restrict__ W0,
                             const float* __restrict__ b0,
                             float* __restrict__ h,
                             int n)
{
    // 128 K-pairs x 64 cols, float2 each = 64 KB
    __shared__ float2 w0s[(2 * DIN / 2) * HID];
    {
        // chunk i: K-pair p = i>>4, col group n4 = (i&15)*4
        for (int i = threadIdx.x; i < (2 * DIN / 2) * (HID / 4); i += blockDim.x) {
            int p  = i >> 4;
            int n4 = (i & 15) * 4;
            float4 r0 = *(const float4*)(W0 + (size_t)(2 * p)     * HID + n4);
            float4 r1 = *(const float4*)(W0 + (size_t)(2 * p + 1) * HID + n4);
            float4* dst4 = (float4*)&w0s[p * HID + n4];
            dst4[0] = make_float4(r0.x, r1.x, r0.y, r1.y);
            dst4[1] = make_float4(r0.z, r1.z, r0.w, r1.w);
        }
    }
    __syncthreads();

    int wave  = threadIdx.x >> 5;
    int lane  = threadIdx.x & 31;
    int tile  = blockIdx.x * (blockDim.x >> 5) + wave;   // 32-node tile
    int tiles = (n + 31) >> 5;
    if (tile >= tiles) return;

    int half = lane >> 4;       // 0: K{0,1} / rows 0-7   1: K{2,3} / rows 8-15
    int m    = lane & 15;       // A row index == B/D column index
    int koff = half * 2;

    int node0 = tile * 32 + m;
    int node1 = tile * 32 + 16 + m;
    if (node0 >= n) node0 = n - 1;      // clamp: keep EXEC all-ones for WMMA
    if (node1 >= n) node1 = n - 1;

    float inv0 = 1.0f / fmaxf(cnt[node0], 1.0f);
    float inv1 = 1.0f / fmaxf(cnt[node1], 1.0f);

    const float* xrow0 = x   + (size_t)node0 * DIN;
    const float* xrow1 = x   + (size_t)node1 * DIN;
    const float* arow0 = agg + (size_t)node0 * DIN;
    const float* arow1 = agg + (size_t)node1 * DIN;

    v8f acc[2][4] = {};

    // K = 0..127 : x half of the concat
    #pragma unroll 4
    for (int kb = 0; kb < DIN; kb += 4) {
        int p = (kb + koff) >> 1;       // kb+koff is even
        v2f a0, a1;
        const float* ap0 = xrow0 + kb + koff;
        const float* ap1 = xrow1 + kb + koff;
        a0.x = ap0[0]; a0.y = ap0[1];
        a1.x = ap1[0]; a1.y = ap1[1];
        const float2* wr = &w0s[p * HID + m];
        #pragma unroll
        for (int t = 0; t < 4; ++t) {
            float2 bp = wr[t * 16];
            v2f b; b.x = bp.x; b.y = bp.y;
            acc[0][t] = __builtin_amdgcn_wmma_f32_16x16x4_f32(
                false, a0, false, b, (short)0, acc[0][t], false, false);
            acc[1][t] = __builtin_amdgcn_wmma_f32_16x16x4_f32(
                false, a1, false, b, (short)0, acc[1][t], false, false);
        }
    }
    // K = 128..255 : mean-aggregated half (divide folded into A load)
    #pragma unroll 4
    for (int kb = 0; kb < DIN; kb += 4) {
        int p = (DIN + kb + koff) >> 1;
        v2f a0, a1;
        const float* ap0 = arow0 + kb + koff;
        const float* ap1 = arow1 + kb + koff;
        a0.x = ap0[0] * inv0; a0.y = ap0[1] * inv0;
        a1.x = ap1[0] * inv1; a1.y = ap1[1] * inv1;
        const float2* wr = &w0s[p * HID + m];
        #pragma unroll
        for (int t = 0; t < 4; ++t) {
            float2 bp = wr[t * 16];
            v2f b; b.x = bp.x; b.y = bp.y;
            acc[0][t] = __builtin_amdgcn_wmma_f32_16x16x4_f32(
                false, a0, false, b, (short)0, acc[0][t], false, false);
            acc[1][t] = __builtin_amdgcn_wmma_f32_16x16x4_f32(
                false, a1, false, b, (short)0, acc[1][t], false, false);
        }
    }

    // Write-out: bias + ReLU.  acc[s][t][v] -> row tile*32 + s*16 + half*8 + v,
    // col t*16 + m.
    #pragma unroll
    for (int sblk = 0; sblk < 2; ++sblk) {
        int row0 = tile * 32 + sblk * 16 + half * 8;
        #pragma unroll
        for (int v = 0; v < 8; ++v) {
            int row = row0 + v;
            if (row < n) {
                float* hr = h + (size_t)row * HID;
                #pragma unroll
                for (int t = 0; t < 4; ++t) {
                    int col = t * 16 + m;
                    hr[col] = fmaxf(acc[sblk][t][v] + b0[col], 0.0f);
                }
            }
        }
    }
}

// ---------------------------------------------------------------------------
// Kernel 3: layer-1 edge scatter.  One wave per edge, float2 per lane (H=64).
// msg = h[src] * act_attr, atomic-added into agg1[dst].
// ---------------------------------------------------------------------------
__global__ void edge_scatter_l1(const float* __restrict__ h,
                                const int*   __restrict__ ei,
                                const float* __restrict__ eattr,
                                float* __restrict__ agg,
                                int E)
{
    int wib  = __builtin_amdgcn_readfirstlane(threadIdx.x >> 5);
    int e    = blockIdx.x * (blockDim.x >> 5) + wib;
    int lane = threadIdx.x & 31;
    if (e >= E) return;
    int s = ei[e];
    int d = ei[(size_t)E + e];
    float2 hv = ((const float2*)(h + (size_t)s * HID))[lane];
    float2 av = ((const float2*)(eattr + (size_t)e * HID))[lane];
    float* a = agg + (size_t)d * HID + lane * 2;
    atomicAdd(a + 0, hv.x * av.x);
    atomicAdd(a + 1, hv.y * av.y);
}

// ---------------------------------------------------------------------------
// Kernel 4: layer-1 output projection.  out = concat([h, agg1/cnt]) @ W1 + b1.
// Dout=2, so a per-thread dot against W1 (cached in LDS) is the right shape.
// ---------------------------------------------------------------------------
__global__ void node_out_l1(const float* __restrict__ h,
                            const float* __restrict__ agg,
                            const float* __restrict__ cnt,
                            const float* __restrict__ W1,
                            const float* __restrict__ b1,
                            float* __restrict__ out,
                            int n)
{
    __shared__ float w[2 * HID * DOUT];   // 256 floats
    __shared__ float bb[DOUT];
    for (int i = threadIdx.x; i < 2 * HID * DOUT; i += blockDim.x) w[i] = W1[i];
    if (threadIdx.x < DOUT) bb[threadIdx.x] = b1[threadIdx.x];
    __syncthreads();

    int i = blockIdx.x * blockDim.x + threadIdx.x;
    if (i >= n) return;

    float inv  = 1.0f / fmaxf(cnt[i], 1.0f);
    float acc0 = bb[0];
    float acc1 = bb[1];
    const float* hr = h   + (size_t)i * HID;
    const float* ar = agg + (size_t)i * HID;
    #pragma unroll 8
    for (int k = 0; k < HID; ++k) {
        float v = hr[k];
        acc0 += v * w[k * DOUT + 0];
        acc1 += v * w[k * DOUT + 1];
    }
    #pragma unroll 8
    for (int k = 0; k < HID; ++k) {
        float v = ar[k] * inv;
        acc0 += v * w[(HID + k) * DOUT + 0];
        acc1 += v * w[(HID + k) * DOUT + 1];
    }
    out[(size_t)i * DOUT + 0] = acc0;
    out[(size_t)i * DOUT + 1] = acc1;
}

// ---------------------------------------------------------------------------
extern "C" void kernel_launch(void* const* d_in, const int* in_sizes, int n_in,
                              void* d_out, int out_size, void* d_ws, size_t ws_size,
                              hipStream_t stream)
{
    const float* x   = (const float*)d_in[0];
    const int*   ei  = (const int*)  d_in[1];
    const float* env = (const float*)d_in[2];
    const float* act = (const float*)d_in[3];
    const float* W0  = (const float*)d_in[4];
    const float* b0  = (const float*)d_in[5];
    const float* W1  = (const float*)d_in[6];
    const float* b1  = (const float*)d_in[7];
    float* out = (float*)d_out;

    const int N = in_sizes[0] / DIN;
    const int E = in_sizes[1] / 2;

    // Workspace carve-up (256B-aligned regions).
    char* ws = (char*)d_ws;
    size_t off = 0;
    auto carve = [&](size_t bytes) {
        char* p = ws + off;
        off = (off + bytes + 255) & ~(size_t)255;
        return p;
    };
    float* agg0 = (float*)carve((size_t)N * DIN * sizeof(float));  // 51.2 MB
    float* cnt  = (float*)carve((size_t)N * sizeof(float));        //  0.4 MB
    float* hbuf = (float*)carve((size_t)N * HID * sizeof(float));  // 25.6 MB
    float* agg1 = (float*)carve((size_t)N * HID * sizeof(float));  // 25.6 MB
    (void)ws_size; (void)n_in; (void)out_size;

    hipMemsetAsync(agg0, 0, (size_t)N * DIN * sizeof(float), stream);
    hipMemsetAsync(cnt,  0, (size_t)N * sizeof(float),       stream);
    hipMemsetAsync(agg1, 0, (size_t)N * HID * sizeof(float), stream);

    // Layer 0
    {
        long long threads = (long long)E * 32;
        int blocks = (int)((threads + 255) / 256);
        edge_scatter_l0<<<blocks, 256, 0, stream>>>(x, ei, env, agg0, cnt, E);
    }
    {
        int tiles  = (N + 31) / 32;
        int blocks = (tiles + 3) / 4;      // 4 waves (128 threads) per block
        node_gemm_l0<<<blocks, 128, 0, stream>>>(x, agg0, cnt, W0, b0, hbuf, N);
    }
    // Layer 1
    {
        long long threads = (long long)E * 32;
        int blocks = (int)((threads + 255) / 256);
        edge_scatter_l1<<<blocks, 256, 0, stream>>>(hbuf, ei, act, agg1, E);
    }
    {
        int blocks = (N + 255) / 256;
        node_out_l1<<<blocks, 256, 0, stream>>>(hbuf, agg1, cnt, W1, b1, out, N);
    }
}